// MultinomialDiffusion_5454608466081
// MI455X (gfx1250) — compile-verified
//
#include <hip/hip_runtime.h>
#include <math.h>

#define KDIM   208
#define NSLICE 16
#define ROWS   16
#define BLOCK  256
#define TSTEPS 1000

// ---- compile-time slice geometry (LEVELS = 2,3,4,5,6,8,10,12,16,20,24,32,16,10,20,20) ----
__constant__ int c_sl_start[NSLICE] = {0,2,5,9,14,20,28,38,50,66,86,110,142,158,168,188};
__constant__ int c_sl_end[NSLICE]   = {2,5,9,14,20,28,38,50,66,86,110,142,158,168,188,208};
__constant__ float c_lognc[NSLICE]  = {
    0.69314718f, 1.09861229f, 1.38629436f, 1.60943791f,
    1.79175947f, 2.07944154f, 2.30258509f, 2.48490665f,
    2.77258872f, 2.99573227f, 3.17805383f, 3.46573590f,
    2.77258872f, 2.30258509f, 2.99573227f, 2.99573227f};

// ---------------- CDNA5 async global->LDS staging (guarded, fallback-safe) ----------------
#if defined(__AMDGCN__) && __has_builtin(__builtin_amdgcn_global_load_async_to_lds_b128)
typedef int v4i __attribute__((vector_size(16)));
typedef __attribute__((address_space(1))) v4i gv4i;
typedef __attribute__((address_space(3))) v4i lv4i;
__device__ __forceinline__ void async_copy16(const float* g, float* l) {
    // one b128 per lane: global -> LDS via the async data path (ASYNCcnt)
    __builtin_amdgcn_global_load_async_to_lds_b128((gv4i*)g, (lv4i*)l, 0, 0);
}
#else
__device__ __forceinline__ void async_copy16(const float* g, float* l) {
    *(float4*)l = *(const float4*)g;
}
#endif

#if defined(__AMDGCN__) && __has_builtin(__builtin_amdgcn_s_wait_asynccnt)
// async loads complete in order per wave: waiting to <=N retires all older ops
#define WAIT_ASYNC(n) __builtin_amdgcn_s_wait_asynccnt(n)
#else
#define WAIT_ASYNC(n) ((void)0)
#endif

// ---------------- helpers ----------------
__device__ __forceinline__ float logaddexpf_(float a, float b) {
    float m = fmaxf(a, b);
    float d = fabsf(a - b);
    return m + __logf(1.0f + __expf(-d));
}

// per-(row,slice) logsumexp over LDS-resident tile; thread = row*16 + slice
__device__ __forceinline__ void slice_lse(const float* __restrict__ s_data,
                                          float* __restrict__ s_lse,
                                          int rowsHere, int tid) {
    int row = tid >> 4;
    int sl  = tid & 15;
    if (row < rowsHere) {
        const float* p = s_data + row * KDIM;
        int a = c_sl_start[sl], b = c_sl_end[sl];
        float m = -3.4e38f;
        for (int k = a; k < b; ++k) m = fmaxf(m, p[k]);
        float s = 0.0f;
        for (int k = a; k < b; ++k) s += __expf(p[k] - m);
        s_lse[tid] = m + __logf(s);
    }
}

// ---------------- setup: LOG_ALPHA_BAR[t] = sum_{i<=t} log(1 - beta_i) into d_ws ----------
__global__ __launch_bounds__(BLOCK)
void setup_lab_kernel(float* __restrict__ lab) {
    __shared__ float part[BLOCK];
    __shared__ float pref[BLOCK];
    const int tid = threadIdx.x;
    float local[4];
    float s = 0.0f;
    #pragma unroll
    for (int j = 0; j < 4; ++j) {
        int i = tid * 4 + j;
        local[j] = 0.0f;
        if (i < TSTEPS) {
            float beta = 1.0e-4f + (float)i * ((0.02f - 1.0e-4f) / 999.0f);
            local[j] = __logf(1.0f - beta);
            s += local[j];
        }
    }
    part[tid] = s;
    __syncthreads();
    if (tid == 0) {
        float acc = 0.0f;
        for (int q = 0; q < BLOCK; ++q) { float p = part[q]; pref[q] = acc; acc += p; }
    }
    __syncthreads();
    float acc = pref[tid];
    #pragma unroll
    for (int j = 0; j < 4; ++j) {
        int i = tid * 4 + j;
        if (i < TSTEPS) { acc += local[j]; lab[i] = acc; }
    }
}

// ---------------- main fused kernel: one HBM pass per array ----------------
__global__ __launch_bounds__(BLOCK)
void md_posterior_kernel(const float* __restrict__ model_pred,
                         const float* __restrict__ log_x_t,
                         const int*   __restrict__ tstep,
                         const float* __restrict__ lab_tab,
                         float* __restrict__ out, int B) {
    __shared__ __align__(16) float s_a[ROWS * KDIM];   // model_pred tile
    __shared__ __align__(16) float s_b[ROWS * KDIM];   // log_x_t tile, then log_tilde in place
    __shared__ float s_lse[ROWS * NSLICE];
    __shared__ float s_sc[ROWS][4];                    // la, loma, lab, lomab
    __shared__ int   s_t0[ROWS];
    __shared__ unsigned char s_sid[KDIM];

    const int tid  = threadIdx.x;
    const int row0 = blockIdx.x * ROWS;
    const int rowsHere = min(ROWS, B - row0);
    const int nv4 = rowsHere * (KDIM / 4);

    const float* gmp = model_pred + (size_t)row0 * KDIM;
    const float* glx = log_x_t    + (size_t)row0 * KDIM;

    // kick off async DMA: tile A (model_pred) first, then tile B (log_x_t)
    for (int i = tid; i < nv4; i += BLOCK) async_copy16(gmp + 4 * i, s_a + 4 * i);
    for (int i = tid; i < nv4; i += BLOCK) async_copy16(glx + 4 * i, s_b + 4 * i);

    // slice id per column (overlaps with DMA)
    if (tid < KDIM) {
        int sid = 0;
        #pragma unroll
        for (int s = 0; s < NSLICE - 1; ++s) sid += (tid >= c_sl_end[s]) ? 1 : 0;
        s_sid[tid] = (unsigned char)sid;
    }
    // per-row timestep scalars (linear beta schedule is closed form; only log(alpha_bar) needs the table)
    if (tid < rowsHere) {
        int tt = tstep[row0 + tid];
        float beta  = 1.0e-4f + (float)tt * ((0.02f - 1.0e-4f) / 999.0f);
        float la    = __logf(1.0f - beta);          // LOG_ALPHAS[t]
        float loma  = __logf(beta + 1e-40f);        // LOG_ONE_MINUS_ALPHAS[t]
        float lab   = 0.0f, lomab = 0.0f;
        int is0 = (tt == 0) ? 1 : 0;
        if (!is0) {
            lab   = lab_tab[tt - 1];                             // LOG_ALPHA_BAR[t-1]
            lomab = __logf(1.0f - __expf(lab) + 1e-40f);         // LOG_ONE_MINUS_ALPHA_BAR[t-1]
        }
        s_sc[tid][0] = la; s_sc[tid][1] = loma; s_sc[tid][2] = lab; s_sc[tid][3] = lomab;
        s_t0[tid] = is0;
    }

    // Each thread issued nA then nB b128s with nA == nB <= 4; in-order completion means
    // asynccnt <= 3 guarantees every tile-A op of this thread retired. Barrier makes it
    // collective: tile A is fully in LDS while tile B may still be streaming.
    WAIT_ASYNC(3);
    __syncthreads();

    // pass 1: per-slice LSE of model_pred (overlaps tile-B DMA)
    slice_lse(s_a, s_lse, rowsHere, tid);

    WAIT_ASYNC(0);          // tile B landed
    __syncthreads();        // + s_lse visible

    // pass 2: fused elementwise -> log_tilde_theta (overwrites s_b in place)
    for (int i = tid; i < nv4; i += BLOCK) {
        float4 a = ((const float4*)s_a)[i];
        float4 b = ((const float4*)s_b)[i];
        int base = i * 4;
        int row  = base / KDIM;          // KDIM % 4 == 0 -> float4 never crosses a row
        int k0   = base - row * KDIM;
        float la = s_sc[row][0], loma = s_sc[row][1], lab = s_sc[row][2], lomab = s_sc[row][3];
        int is0 = s_t0[row];
        float va[4] = {a.x, a.y, a.z, a.w};
        float vb[4] = {b.x, b.y, b.z, b.w};
        float r[4];
        #pragma unroll
        for (int e = 0; e < 4; ++e) {
            int sid = s_sid[k0 + e];
            float lognc = c_lognc[sid];
            float lhat  = va[e] - s_lse[row * NSLICE + sid];     // log_hat_x0
            float lp    = is0 ? lhat : logaddexpf_(lab + lhat, lomab - lognc);
            float os    = logaddexpf_(la + vb[e], loma - lognc); // one_step
            r[e] = lp + os;
        }
        ((float4*)s_b)[i] = make_float4(r[0], r[1], r[2], r[3]);
    }
    __syncthreads();

    // pass 3: per-slice LSE of log_tilde_theta
    slice_lse(s_b, s_lse, rowsHere, tid);
    __syncthreads();

    // pass 4: normalize + coalesced b128 store
    float* gout = out + (size_t)row0 * KDIM;
    for (int i = tid; i < nv4; i += BLOCK) {
        float4 b = ((const float4*)s_b)[i];
        int base = i * 4;
        int row  = base / KDIM;
        int k0   = base - row * KDIM;
        float r[4] = {b.x, b.y, b.z, b.w};
        #pragma unroll
        for (int e = 0; e < 4; ++e)
            r[e] -= s_lse[row * NSLICE + s_sid[k0 + e]];
        ((float4*)gout)[i] = make_float4(r[0], r[1], r[2], r[3]);
    }
}

extern "C" void kernel_launch(void* const* d_in, const int* in_sizes, int n_in,
                              void* d_out, int out_size, void* d_ws, size_t ws_size,
                              hipStream_t stream) {
    const float* model_pred = (const float*)d_in[0];
    const float* log_x_t    = (const float*)d_in[1];
    const int*   tstep      = (const int*)d_in[2];
    float* out = (float*)d_out;
    int B = in_sizes[2];                 // one timestep per row
    float* lab_tab = (float*)d_ws;       // needs TSTEPS*4 = 4000 bytes of scratch

    hipLaunchKernelGGL(setup_lab_kernel, dim3(1), dim3(BLOCK), 0, stream, lab_tab);
    int grid = (B + ROWS - 1) / ROWS;
    hipLaunchKernelGGL(md_posterior_kernel, dim3(grid), dim3(BLOCK), 0, stream,
                       model_pred, log_x_t, tstep, lab_tab, out, B);
}